// SignatureGCN_37014028157315
// MI455X (gfx1250) — compile-verified
//
#include <hip/hip_runtime.h>
#include <hip/hip_bf16.h>

// ---------------------------------------------------------------------------
// GCN forward for MI455X (gfx1250, wave32).
// GEMMs: V_WMMA_F32_16X16X4_F32, one 16x16 tile per wave, K stepped by 4.
// K/NPAD/NC are template constants -> LDS reads use immediate ds offsets,
// A reads use immediate global offsets, ragged guards constant-fold.
// Edge aggregation: hardware f32 global atomics (unsafeAtomicAdd).
// ---------------------------------------------------------------------------

typedef float v2f __attribute__((ext_vector_type(2)));
typedef float v8f __attribute__((ext_vector_type(8)));

#define WAVES_PER_BLOCK 8
#define TBLOCK (WAVES_PER_BLOCK * 32)

__device__ __forceinline__ void atomic_add_f32(float* p, float v) {
#if defined(__HIP_PLATFORM_AMD__) || defined(__AMDGCN__)
  unsafeAtomicAdd(p, v);   // lowers to global_atomic_add_f32
#else
  atomicAdd(p, v);
#endif
}

__host__ __device__ constexpr int log2c(int v) { return v <= 1 ? 0 : 1 + log2c(v >> 1); }

// ---- degree / dinv --------------------------------------------------------
__global__ void k_fill(float* __restrict__ p, float val, int n) {
  int i = blockIdx.x * blockDim.x + threadIdx.x;
  if (i < n) p[i] = val;
}

__global__ void k_deg_accum(const int* __restrict__ dst, float* __restrict__ deg, int e) {
  int i = blockIdx.x * blockDim.x + threadIdx.x;
  if (i < e) atomic_add_f32(&deg[dst[i]], 1.0f);
}

__global__ void k_rsqrt_inplace(float* __restrict__ p, int n) {
  int i = blockIdx.x * blockDim.x + threadIdx.x;
  if (i < n) {
    float d = p[i];
    p[i] = (d > 0.0f) ? __frsqrt_rn(d) : 0.0f;
  }
}

// ---- f32 WMMA GEMM:  C[M,NC] = A[M,K] @ B[K,NC] (+ bias) ------------------
// One wave computes one 16x16 tile; M % 16 == 0, K % 4 == 0, NPAD = ntn*16.
// B staged into LDS zero-padded to NPAD cols: inner loop has no predication
// (EXEC all-1s at every v_wmma) and all LDS/global offsets are immediates.
template <int K, int NPAD, int NC>
__global__ __launch_bounds__(TBLOCK)
void k_wmma_gemm_f32(const float* __restrict__ A,
                     const float* __restrict__ B,
                     const float* __restrict__ bias,   // may be nullptr
                     float* __restrict__ C,
                     int Mrows) {
  constexpr int NTN   = NPAD / 16;
  constexpr int LOGNP = log2c(NPAD);
  __shared__ float ldsB[K * NPAD];

  // cooperative, zero-padded copy of B into LDS (coalesced over columns)
#pragma unroll 4
  for (int idx = threadIdx.x; idx < K * NPAD; idx += TBLOCK) {
    const int kk = idx >> LOGNP;
    const int nn = idx & (NPAD - 1);
    ldsB[idx] = (nn < NC) ? B[kk * NC + nn] : 0.0f;   // guard folds if NC==NPAD
  }
  __syncthreads();

  const int lane = threadIdx.x & 31;
  const int wave = threadIdx.x >> 5;
  const int gid  = blockIdx.x * WAVES_PER_BLOCK + wave;   // wave-uniform
  const int tm   = gid / NTN;
  const int tn   = gid - tm * NTN;
  if (tm * 16 >= Mrows) return;                            // wave-uniform exit

  const int half = lane >> 4;       // 0: lanes 0-15, 1: lanes 16-31
  const int l15  = lane & 15;
  const int arow = tm * 16 + l15;   // A layout: row = lane&15 in both halves
  const int bcol = tn * 16 + l15;   // B/C/D: col = lane&15
  const int koff = half * 2;        // K pair 0-1 (half 0) vs 2-3 (half 1)

  const float* __restrict__ Arow = A + (size_t)arow * K + koff;
  const float* __restrict__ Bl   = ldsB + (koff << LOGNP) + bcol;

  v8f acc = {};
#pragma unroll 8
  for (int k = 0; k < K; k += 4) {
    v2f a, b;
    a.x = Arow[k];                   // global_load_b64, immediate offset
    a.y = Arow[k + 1];
    b.x = Bl[(k)     << LOGNP];      // ds_load, immediate offset
    b.y = Bl[(k + 1) << LOGNP];
    // (neg_a, A, neg_b, B, c_mod, C, reuse_a, reuse_b)
    acc = __builtin_amdgcn_wmma_f32_16x16x4_f32(
        false, a, false, b, (short)0, acc, false, false);
  }

  if (bcol < NC) {                   // folds away unless ragged (NC < NPAD)
    const float bb = bias ? bias[bcol] : 0.0f;
    const int rbase = tm * 16 + half * 8;   // C/D: VGPR r -> row r + 8*half
#pragma unroll
    for (int r = 0; r < 8; ++r)
      C[(size_t)(rbase + r) * NC + bcol] = acc[r] + bb;
  }
}

// ---- aggregation ----------------------------------------------------------
// agg[i,f] = dinv[i]^2 * h[i,f]   (self-loop term; F is a power of two)
__global__ void k_selfloop_init(const float* __restrict__ h,
                                const float* __restrict__ dinv,
                                float* __restrict__ agg,
                                long long total, int logF) {
  long long idx = (long long)blockIdx.x * blockDim.x + threadIdx.x;
  if (idx >= total) return;
  int i = (int)(idx >> logF);
  float di = dinv[i];
  agg[idx] = di * di * h[idx];
}

// agg[dst,f] += dinv[src]*dinv[dst] * h[src,f]  over all edges
__global__ void k_scatter_edges(const float* __restrict__ h,
                                const int* __restrict__ src,
                                const int* __restrict__ dst,
                                const float* __restrict__ dinv,
                                float* __restrict__ agg,
                                long long total, int F, int logF) {
  long long idx = (long long)blockIdx.x * blockDim.x + threadIdx.x;
  if (idx >= total) return;
  int e = (int)(idx >> logF);
  int f = (int)(idx & (F - 1));
  int s = src[e], d = dst[e];
  float coef = dinv[s] * dinv[d];
  atomic_add_f32(&agg[(size_t)d * F + f], h[(size_t)s * F + f] * coef);
}

// h[i,f] = relu(agg[i,f] + b[f])   (in place on agg)
__global__ void k_bias_relu(float* __restrict__ agg,
                            const float* __restrict__ b,
                            long long total, int F) {
  long long idx = (long long)blockIdx.x * blockDim.x + threadIdx.x;
  if (idx >= total) return;
  int f = (int)(idx % F);
  float v = agg[idx] + b[f];
  agg[idx] = v > 0.0f ? v : 0.0f;
}

// ---------------------------------------------------------------------------
extern "C" void kernel_launch(void* const* d_in, const int* in_sizes, int n_in,
                              void* d_out, int out_size, void* d_ws, size_t ws_size,
                              hipStream_t stream) {
  const int F_IN = 128, H1 = 128, H2 = 64;
  const float* x  = (const float*)d_in[0];
  const int*   ei = (const int*)d_in[1];      // int32 [2, E] row-major
  const float* W1 = (const float*)d_in[2];
  const float* b1 = (const float*)d_in[3];
  const float* W2 = (const float*)d_in[4];
  const float* b2 = (const float*)d_in[5];
  const float* Wh = (const float*)d_in[6];
  const float* bh = (const float*)d_in[7];
  float* out = (float*)d_out;

  const int N = in_sizes[0] / F_IN;           // 50000
  const int E = in_sizes[1] / 2;              // 625000
  const int* src = ei;
  const int* dst = ei + E;
  (void)out_size; (void)ws_size; (void)n_in;

  // workspace layout
  float* dinv = (float*)d_ws;                 // N  (holds deg, then rsqrt)
  float* h1   = dinv + N;                     // N*H1
  float* agg1 = h1 + (size_t)N * H1;          // N*H1
  float* h2   = agg1 + (size_t)N * H1;        // N*H2
  float* agg2 = h2 + (size_t)N * H2;          // N*H2

  const int TB = 256;
  // 1) degrees (self-loop contributes 1 to every node), then dinv = rsqrt
  k_fill<<<(N + TB - 1) / TB, TB, 0, stream>>>(dinv, 1.0f, N);
  k_deg_accum<<<(E + TB - 1) / TB, TB, 0, stream>>>(dst, dinv, E);
  k_rsqrt_inplace<<<(N + TB - 1) / TB, TB, 0, stream>>>(dinv, N);

  const int mtiles = (N + 15) / 16;
  auto gemm_blocks = [&](int ntn) {
    long long tiles = (long long)mtiles * ntn;
    return (int)((tiles + WAVES_PER_BLOCK - 1) / WAVES_PER_BLOCK);
  };

  // 2) h1 = x @ W1                                  (K=128, N=128)
  k_wmma_gemm_f32<128, 128, 128><<<gemm_blocks(8), TBLOCK, 0, stream>>>(
      x, W1, nullptr, h1, N);

  // 3) agg1 = D^-1/2 A_hat D^-1/2 h1 ; relu(+b1)
  {
    long long tot = (long long)N * H1;
    k_selfloop_init<<<(int)((tot + TB - 1) / TB), TB, 0, stream>>>(h1, dinv, agg1, tot, 7);
    long long et = (long long)E * H1;
    k_scatter_edges<<<(int)((et + TB - 1) / TB), TB, 0, stream>>>(
        h1, src, dst, dinv, agg1, et, H1, 7);
    k_bias_relu<<<(int)((tot + TB - 1) / TB), TB, 0, stream>>>(agg1, b1, tot, H1);
  }

  // 4) h2 = agg1 @ W2                               (K=128, N=64)
  k_wmma_gemm_f32<128, 64, 64><<<gemm_blocks(4), TBLOCK, 0, stream>>>(
      agg1, W2, nullptr, h2, N);

  // 5) agg2 = D^-1/2 A_hat D^-1/2 h2 ; relu(+b2)
  {
    long long tot = (long long)N * H2;
    k_selfloop_init<<<(int)((tot + TB - 1) / TB), TB, 0, stream>>>(h2, dinv, agg2, tot, 6);
    long long et = (long long)E * H2;
    k_scatter_edges<<<(int)((et + TB - 1) / TB), TB, 0, stream>>>(
        h2, src, dst, dinv, agg2, et, H2, 6);
    k_bias_relu<<<(int)((tot + TB - 1) / TB), TB, 0, stream>>>(agg2, b2, tot, H2);
  }

  // 6) out = agg2 @ Wh + bh                         (K=64, N=50 ragged -> pad 64)
  k_wmma_gemm_f32<64, 64, 50><<<gemm_blocks(4), TBLOCK, 0, stream>>>(
      agg2, Wh, bh, out, N);
}